// CRFLoss_12601434046728
// MI455X (gfx1250) — compile-verified
//
#include <hip/hip_runtime.h>
#include <hip/hip_bf16.h>
#include <math.h>

#define B_DIM 512
#define T_DIM 1024
#define L_DIM 48

#define INV_LN2 1.44269504088896340736f  // log2(e)
#define LN2     0.69314718055994530942f  // ln(2)

typedef __attribute__((ext_vector_type(2))) float v2f;
typedef __attribute__((ext_vector_type(8))) float v8f;

// ---------------------------------------------------------------------------
// Forward algorithm.  One workgroup = 3 waves = 96 threads handles 16 batch
// rows through all T steps.  Wave w owns output k-tile w (columns w*16..+15).
// State alpha2 = alpha/ln2 is kept in base-2 units so the per-step
// transcendentals are raw v_exp_f32 / v_log_f32 (no libm denorm guards:
// D >= 0.9 always since P[argmax]=1 and E in [0.905, 1.105]).
// Per step/wave: row-max (LDS), 8 v_exp, 12 chained V_WMMA_F32_16X16X4_F32,
// 8 v_log + 8 one-step-pipelined emission loads, LDS writeback. 2 barriers.
// ---------------------------------------------------------------------------
__global__ __launch_bounds__(96) void crf_forward_kernel(
    const float* __restrict__ scores,   // [B, T, L]
    const float* __restrict__ start,    // [L]
    const float* __restrict__ Tm,       // [L, L]  Tm[k*L + j]
    const float* __restrict__ endv,     // [L]
    float* __restrict__ norm_out)       // [B]
{
    __shared__ __align__(16) float Eb[L_DIM * L_DIM];   // Eb[j*L + k] = exp(Tm[k*L + j])
    __shared__ __align__(16) float alphaS[16 * L_DIM];  // alpha2[b][j]  (base-2 units)
    __shared__ __align__(16) float PS[16 * L_DIM];      // 2^(alpha2 - m2)
    __shared__ float mS[16];

    const int tid  = threadIdx.x;
    const int lane = tid & 31;
    const int wave = tid >> 5;         // 0..2, uniform per wave: owns k-tile `wave`
    const int base = blockIdx.x * 16;  // global batch base
    const int b    = lane & 15;        // A-layout M / D-layout N index
    const int half = lane >> 4;
    const int koff = half * 2;         // A/B fragment K offset per lane half

    // --- E = exp(Tm^T), computed once per block (linear domain, unchanged) ---
    for (int i = tid; i < L_DIM * L_DIM; i += 96) {
        int j = i / L_DIM, k = i % L_DIM;
        Eb[i] = __expf(Tm[k * L_DIM + j]);
    }
    __syncthreads();

    // --- this wave's 12 B fragments (Eᵀ, columns wave*16..+15) ---
    // B 4x16 layout: lane holds N = lane%16, K = (lane/16)*2 + vgpr
    v2f bfrag[12];
#pragma unroll
    for (int cj = 0; cj < 12; ++cj) {
        int j0 = cj * 4 + koff;
        v2f t;
        t.x = Eb[(j0    ) * L_DIM + wave * 16 + b];
        t.y = Eb[(j0 + 1) * L_DIM + wave * 16 + b];
        bfrag[cj] = t;
    }

    // --- alpha2(t=0) = (start + scores[:,0,:]) / ln2 ---
    for (int i = tid; i < 16 * L_DIM; i += 96) {
        int bb = i / L_DIM, j = i % L_DIM;
        alphaS[i] = (start[j] + scores[(size_t)(base + bb) * T_DIM * L_DIM + j]) * INV_LN2;
    }

    // --- prefetch emissions for t=1 (this wave's k-tile columns) ---
    float snext[8];
#pragma unroll
    for (int v = 0; v < 8; ++v) {
        int bb = half * 8 + v;
        snext[v] = scores[((size_t)(base + bb) * T_DIM + 1) * L_DIM + wave * 16 + b];
    }

    for (int t = 1; t < T_DIM; ++t) {
        __syncthreads();  // alphaS ready (init or previous writeback)

        // m2[b] = max_j alpha2[b][j]  (redundant across the 3 waves; no extra barrier)
        float m = -INFINITY;
        const float4* row = (const float4*)&alphaS[b * L_DIM];
#pragma unroll
        for (int q = 0; q < 12; ++q) {
            float4 v4 = row[q];
            m = fmaxf(m, fmaxf(fmaxf(v4.x, v4.y), fmaxf(v4.z, v4.w)));
        }
        if (tid < 16) mS[tid] = m;

        // P[b][j] = 2^(alpha2 - m2) for this wave's 16 j's (8 per lane half)
        {
            int j0 = wave * 16 + half * 8;
#pragma unroll
            for (int jj = 0; jj < 8; jj += 2) {
                float2 p;
                p.x = __builtin_amdgcn_exp2f(alphaS[b * L_DIM + j0 + jj    ] - m);
                p.y = __builtin_amdgcn_exp2f(alphaS[b * L_DIM + j0 + jj + 1] - m);
                *(float2*)&PS[b * L_DIM + j0 + jj] = p;
            }
        }
        __syncthreads();  // PS + mS ready

        // consume step-t emissions loaded last iteration; issue loads for t+1
        float scur[8];
#pragma unroll
        for (int v = 0; v < 8; ++v) scur[v] = snext[v];
        if (t + 1 < T_DIM) {
#pragma unroll
            for (int v = 0; v < 8; ++v) {
                int bb = half * 8 + v;
                snext[v] = scores[((size_t)(base + bb) * T_DIM + (t + 1)) * L_DIM
                                  + wave * 16 + b];
            }
        }

        // D-tile = P(16x48) x Eᵀ(48x16) : 12 chained WMMA f32 16x16x4
        v8f c = {};
#pragma unroll
        for (int cj = 0; cj < 12; ++cj) {
            // A 16x4: lane holds M = lane%16, K = cj*4 + (lane/16)*2 + {0,1}
            v2f a = *(const v2f*)&PS[b * L_DIM + cj * 4 + koff];
            c = __builtin_amdgcn_wmma_f32_16x16x4_f32(false, a, false, bfrag[cj],
                                                      (short)0, c, false, false);
        }

        // alpha2[bb][k] = m2[bb] + log2(D) + scores[bb,t,k]/ln2
        // D layout: lane holds N = lane%16 (=b), M = v + 8*(lane/16)
#pragma unroll
        for (int v = 0; v < 8; ++v) {
            int bb = half * 8 + v;
            float lg = __builtin_amdgcn_logf(c[v]);   // raw v_log_f32 (D >= 0.9)
            alphaS[bb * L_DIM + wave * 16 + b] = fmaf(scur[v], INV_LN2, mS[bb] + lg);
        }
    }
    __syncthreads();

    // normalizer[b] = ln2 * log2sumexp2_k(alpha2[b][k] + end[k]/ln2)
    if (tid < 16) {
        float mm = -INFINITY;
        for (int k = 0; k < L_DIM; ++k)
            mm = fmaxf(mm, fmaf(endv[k], INV_LN2, alphaS[tid * L_DIM + k]));
        float s = 0.f;
        for (int k = 0; k < L_DIM; ++k)
            s += __builtin_amdgcn_exp2f(fmaf(endv[k], INV_LN2, alphaS[tid * L_DIM + k]) - mm);
        norm_out[base + tid] = LN2 * (mm + __builtin_amdgcn_logf(s));
    }
}

// ---------------------------------------------------------------------------
// Gold path score per batch: emissions gather + transition gather + start/end
// ---------------------------------------------------------------------------
__global__ __launch_bounds__(256) void crf_gold_kernel(
    const float* __restrict__ scores, const int* __restrict__ targets,
    const float* __restrict__ start, const float* __restrict__ Tm,
    const float* __restrict__ endv, float* __restrict__ gold_out)
{
    __shared__ float red[256];
    const int bidx = blockIdx.x;
    const int tid  = threadIdx.x;
    const int* tg  = targets + (size_t)bidx * T_DIM;

    float acc = 0.f;
    for (int t = tid; t < T_DIM; t += 256) {
        int cur = tg[t];
        acc += scores[((size_t)bidx * T_DIM + t) * L_DIM + cur];
        if (t > 0) acc += Tm[cur * L_DIM + tg[t - 1]];
    }
    red[tid] = acc;
    __syncthreads();
    for (int s = 128; s > 0; s >>= 1) {
        if (tid < s) red[tid] += red[tid + s];
        __syncthreads();
    }
    if (tid == 0)
        gold_out[bidx] = red[0] + start[tg[0]] + endv[tg[T_DIM - 1]];
}

// ---------------------------------------------------------------------------
// Final mean over batches: out = mean(norm - gold)
// ---------------------------------------------------------------------------
__global__ __launch_bounds__(512) void crf_final_kernel(
    const float* __restrict__ norm, const float* __restrict__ gold,
    float* __restrict__ out)
{
    __shared__ float red[512];
    const int tid = threadIdx.x;
    red[tid] = norm[tid] - gold[tid];
    __syncthreads();
    for (int s = 256; s > 0; s >>= 1) {
        if (tid < s) red[tid] += red[tid + s];
        __syncthreads();
    }
    if (tid == 0) out[0] = red[0] / (float)B_DIM;
}

extern "C" void kernel_launch(void* const* d_in, const int* in_sizes, int n_in,
                              void* d_out, int out_size, void* d_ws, size_t ws_size,
                              hipStream_t stream) {
    (void)in_sizes; (void)n_in; (void)out_size; (void)ws_size;
    const float* scores  = (const float*)d_in[0];
    const int*   targets = (const int*)d_in[1];
    const float* start   = (const float*)d_in[2];
    const float* Tm      = (const float*)d_in[3];
    const float* endv    = (const float*)d_in[4];
    float* out  = (float*)d_out;
    float* norm = (float*)d_ws;          // [512]
    float* gold = norm + B_DIM;          // [512]

    crf_forward_kernel<<<B_DIM / 16, 96, 0, stream>>>(scores, start, Tm, endv, norm);
    crf_gold_kernel<<<B_DIM, 256, 0, stream>>>(scores, targets, start, Tm, endv, gold);
    crf_final_kernel<<<1, 512, 0, stream>>>(norm, gold, out);
}